// SelfAttentionHead_27247272526054
// MI455X (gfx1250) — compile-verified
//
#include <hip/hip_runtime.h>
#include <math.h>

typedef float v2f __attribute__((ext_vector_type(2)));
typedef float v8f __attribute__((ext_vector_type(8)));

namespace {
constexpr int kB  = 8;
constexpr int kT  = 2048;
constexpr int kD  = 1024;
constexpr int kHD = 64;          // head dim d
constexpr int kM  = 64;          // memory tokens
constexpr int kS  = kT + kM;     // 2112 keys
constexpr float kLnEps = 1e-5f;
constexpr float kScale = 0.125f; // d^-0.5 = 1/8
}

// D = A(16x4 f32) * B(4x16 f32) + C(16x16 f32)
// A layout: elem(m,k) -> lane = m + 16*(k/2), vgpr = k%2  (float2 per lane)
// B layout (symmetric): elem(k,n) -> lane = n + 16*(k/2), vgpr = k%2
// C/D layout: elem(m,n) -> lane = n + 16*(m/8), vgpr = m%8
__device__ __forceinline__ v8f wmma_f32(v2f a, v2f b, v8f c) {
  return __builtin_amdgcn_wmma_f32_16x16x4_f32(
      /*neg_a=*/false, a, /*neg_b=*/false, b,
      /*c_mod=*/(short)0, c, /*reuse_a=*/false, /*reuse_b=*/false);
}

// ---------------------------------------------------------------------------
// Kernel A: layernorm(c_emb) -> rows [T, T+M) of k_buf and v_buf, per batch.
// ---------------------------------------------------------------------------
__global__ __launch_bounds__(64)
void ln_mem_kernel(const float* __restrict__ c_emb,
                   const float* __restrict__ gamma,
                   const float* __restrict__ beta,
                   float* __restrict__ k_buf,
                   float* __restrict__ v_buf) {
  const int b = blockIdx.x;
  const int r = threadIdx.x;                // 0..63
  const float* row = c_emb + r * kHD;
  float mu = 0.f;
  #pragma unroll
  for (int i = 0; i < kHD; ++i) mu += row[i];
  mu *= (1.f / kHD);
  float var = 0.f;
  #pragma unroll
  for (int i = 0; i < kHD; ++i) { float d = row[i] - mu; var += d * d; }
  var *= (1.f / kHD);
  const float rstd = rsqrtf(var + kLnEps);
  float* kd = k_buf + ((size_t)b * kS + kT + r) * kHD;
  float* vd = v_buf + ((size_t)b * kS + kT + r) * kHD;
  #pragma unroll
  for (int i = 0; i < kHD; ++i) {
    float y = (row[i] - mu) * rstd * gamma[i] + beta[i];
    kd[i] = y;
    vd[i] = y;
  }
}

// ---------------------------------------------------------------------------
// Kernel B: fused QKV projection.  [B*T x 1024] @ [1024 x (64|64|64)].
// 128 blocks x 256 threads (8 waves). Block tile: 128 rows x 192 cols.
// Wave: 16 rows x 192 cols -> 12 f32 16x16 accumulators.
// Software-pipelined: global loads for K-tile t+1 issued before computing t.
// W staged TRANSPOSED in LDS so B-fragments are contiguous ds_load_b64.
// ---------------------------------------------------------------------------
__global__ __launch_bounds__(256)
void qkv_kernel(const float* __restrict__ x,
                const float* __restrict__ Wq,
                const float* __restrict__ Wk,
                const float* __restrict__ Wv,
                float* __restrict__ q_buf,
                float* __restrict__ k_buf,
                float* __restrict__ v_buf) {
  __shared__ float xs[128][36];    // x tile, row-major (A-frags contiguous)
  __shared__ float wst[192][34];   // W tile transposed: [col][k]

  const int tid  = threadIdx.x;
  const int wid  = tid >> 5;
  const int lane = tid & 31;
  const int half = lane >> 4;
  const int lm   = lane & 15;
  const int r0   = blockIdx.x * 128;
  const int m0   = wid * 16;

  v8f acc[12] = {};
  float4 xreg[4];
  float4 wreg[6];

  auto gload = [&](int kt) {
    #pragma unroll
    for (int i = 0; i < 4; ++i) {           // 128x32 floats
      int idx = tid + 256 * i;
      int row = idx >> 3;
      int c4  = (idx & 7) * 4;
      xreg[i] = *(const float4*)(x + (size_t)(r0 + row) * kD + kt + c4);
    }
    #pragma unroll
    for (int i = 0; i < 6; ++i) {           // 32x192 floats
      int idx = tid + 256 * i;
      int kr  = idx / 48;
      int c4  = idx % 48;
      int mat = c4 >> 4;
      int cc  = (c4 & 15) * 4;
      const float* src = (mat == 0) ? Wq : ((mat == 1) ? Wk : Wv);
      wreg[i] = *(const float4*)(src + (size_t)(kt + kr) * kHD + cc);
    }
  };
  auto lstore = [&]() {
    #pragma unroll
    for (int i = 0; i < 4; ++i) {
      int idx = tid + 256 * i;
      int row = idx >> 3;
      int c4  = (idx & 7) * 4;
      *(float4*)&xs[row][c4] = xreg[i];
    }
    #pragma unroll
    for (int i = 0; i < 6; ++i) {
      int idx = tid + 256 * i;
      int kr  = idx / 48;
      int c4  = idx % 48;
      int col = (c4 >> 4) * 64 + (c4 & 15) * 4;
      wst[col + 0][kr] = wreg[i].x;
      wst[col + 1][kr] = wreg[i].y;
      wst[col + 2][kr] = wreg[i].z;
      wst[col + 3][kr] = wreg[i].w;
    }
  };

  gload(0);
  lstore();
  __syncthreads();

  constexpr int kNT = kD / 32;              // 32 K-tiles
  for (int t = 0; t < kNT; ++t) {
    if (t + 1 < kNT) gload(32 * (t + 1));   // prefetch next tile (no wait yet)
    #pragma unroll
    for (int kk = 0; kk < 8; ++kk) {
      const int k0 = kk * 4;
      v2f a = *(const v2f*)&xs[m0 + lm][k0 + 2 * half];
      #pragma unroll
      for (int c = 0; c < 12; ++c) {
        v2f bf = *(const v2f*)&wst[c * 16 + lm][k0 + 2 * half];
        acc[c] = wmma_f32(a, bf, acc[c]);
      }
    }
    __syncthreads();                        // everyone done reading tile t
    if (t + 1 < kNT) lstore();              // waits on global loads here
    __syncthreads();
  }

  // scatter D-layout accumulators to q/k/v buffers
  #pragma unroll
  for (int c = 0; c < 12; ++c) {
    const int mat = c >> 2;                 // 0:q 1:k 2:v
    const int hd  = (c & 3) * 16 + lm;
    #pragma unroll
    for (int i = 0; i < 8; ++i) {
      const int gr = r0 + m0 + i + 8 * half;       // row in [0, B*T)
      const float val = acc[c][i];
      if (mat == 0) {
        q_buf[(size_t)gr * kHD + hd] = val;
      } else {
        const int bb = gr / kT;
        const int tt = gr - bb * kT;
        float* dst = (mat == 1) ? k_buf : v_buf;
        dst[((size_t)bb * kS + tt) * kHD + hd] = val;
      }
    }
  }
}

// ---------------------------------------------------------------------------
// Kernel C: flash-style attention. grid (T/128, B), 256 threads (8 waves).
// Block: 128 queries of one batch; wave: 16 queries, full d=64.
// Stream over 33 tiles of 64 keys; online softmax; V staged transposed so
// P@V B-fragments are contiguous ds_load_b64. Pipelined k/v tile loads.
// ---------------------------------------------------------------------------
__global__ __launch_bounds__(256)
void attn_kernel(const float* __restrict__ q_buf,
                 const float* __restrict__ k_buf,
                 const float* __restrict__ v_buf,
                 float* __restrict__ out) {
  __shared__ float qs[128][68];      // queries x dims (pre-scaled by 1/8)
  __shared__ float ks[64][68];       // keys x dims (row-major; S B-frags contiguous)
  __shared__ float vst[64][68];      // TRANSPOSED: [dim][key]
  __shared__ float ps[8][16][68];    // per-wave P scratch (16 q x 64 keys)

  const int tid  = threadIdx.x;
  const int wid  = tid >> 5;
  const int lane = tid & 31;
  const int half = lane >> 4;
  const int lm   = lane & 15;
  const int b    = blockIdx.y;
  const int q0   = blockIdx.x * 128;
  const int m0   = wid * 16;

  float4 kreg[4], vreg[4];
  auto gload = [&](int s0) {
    #pragma unroll
    for (int i = 0; i < 4; ++i) {           // 64x64 floats each
      int idx = tid + 256 * i;
      int row = idx >> 4;
      int c4  = (idx & 15) * 4;
      kreg[i] = *(const float4*)(k_buf + ((size_t)b * kS + s0 + row) * kHD + c4);
      vreg[i] = *(const float4*)(v_buf + ((size_t)b * kS + s0 + row) * kHD + c4);
    }
  };
  auto lstore = [&]() {
    #pragma unroll
    for (int i = 0; i < 4; ++i) {
      int idx = tid + 256 * i;
      int row = idx >> 4;
      int c4  = (idx & 15) * 4;
      *(float4*)&ks[row][c4] = kreg[i];
      vst[c4 + 0][row] = vreg[i].x;
      vst[c4 + 1][row] = vreg[i].y;
      vst[c4 + 2][row] = vreg[i].z;
      vst[c4 + 3][row] = vreg[i].w;
    }
  };

  // stage q tile once, folding in the 1/sqrt(d) scale
  #pragma unroll
  for (int i = 0; i < 8; ++i) {
    int idx = tid + 256 * i;
    int row = idx >> 4;
    int c4  = (idx & 15) * 4;
    float4 v = *(const float4*)(q_buf + ((size_t)b * kT + q0 + row) * kHD + c4);
    v.x *= kScale; v.y *= kScale; v.z *= kScale; v.w *= kScale;
    *(float4*)&qs[row][c4] = v;
  }
  gload(0);
  lstore();
  __syncthreads();

  v8f oacc[4] = {};
  float mrun[8], lrun[8];
  #pragma unroll
  for (int i = 0; i < 8; ++i) { mrun[i] = -INFINITY; lrun[i] = 0.f; }

  constexpr int kNT = kS / 64;              // 33 key tiles
  for (int it = 0; it < kNT; ++it) {
    if (it + 1 < kNT) gload(64 * (it + 1)); // prefetch next k/v tile

    // S = (q*scale) @ k^T : 16 queries x 64 keys, K = 64
    v8f sacc[4] = {};
    #pragma unroll
    for (int kk = 0; kk < 16; ++kk) {
      const int k0 = kk * 4;
      v2f a = *(const v2f*)&qs[m0 + lm][k0 + 2 * half];
      #pragma unroll
      for (int j = 0; j < 4; ++j) {
        v2f bf = *(const v2f*)&ks[j * 16 + lm][k0 + 2 * half];
        sacc[j] = wmma_f32(a, bf, sacc[j]);
      }
    }

    // online softmax. Lane in half h owns stats for rows 8h..8h+7.
    float mnew[8], alpha[8], rsum[8];
    #pragma unroll
    for (int i = 0; i < 8; ++i) {
      float v = fmaxf(fmaxf(sacc[0][i], sacc[1][i]),
                      fmaxf(sacc[2][i], sacc[3][i]));
      #pragma unroll
      for (int off = 1; off < 16; off <<= 1)
        v = fmaxf(v, __shfl_xor(v, off));   // stays within 16-lane half
      mnew[i]  = fmaxf(mrun[i], v);
      alpha[i] = __expf(mrun[i] - mnew[i]);
      mrun[i]  = mnew[i];
      rsum[i]  = 0.f;
    }
    #pragma unroll
    for (int j = 0; j < 4; ++j) {
      #pragma unroll
      for (int i = 0; i < 8; ++i) {
        float p = __expf(sacc[j][i] - mnew[i]);
        sacc[j][i] = p;
        rsum[i] += p;
      }
    }
    #pragma unroll
    for (int i = 0; i < 8; ++i) {
      float rs = rsum[i];
      #pragma unroll
      for (int off = 1; off < 16; off <<= 1)
        rs += __shfl_xor(rs, off);
      lrun[i] = lrun[i] * alpha[i] + rs;
    }
    #pragma unroll
    for (int c = 0; c < 4; ++c)
      #pragma unroll
      for (int i = 0; i < 8; ++i)
        oacc[c][i] *= alpha[i];

    // D-layout -> A-layout bounce through per-wave LDS scratch
    #pragma unroll
    for (int j = 0; j < 4; ++j)
      #pragma unroll
      for (int i = 0; i < 8; ++i)
        ps[wid][i + 8 * half][j * 16 + lm] = sacc[j][i];

    // out += P @ V : K = 64 keys, N = 64 dims (V transposed in LDS)
    #pragma unroll
    for (int kk = 0; kk < 16; ++kk) {
      const int k0 = kk * 4;
      v2f a = *(const v2f*)&ps[wid][lm][k0 + 2 * half];
      #pragma unroll
      for (int c = 0; c < 4; ++c) {
        v2f bf = *(const v2f*)&vst[c * 16 + lm][k0 + 2 * half];
        oacc[c] = wmma_f32(a, bf, oacc[c]);
      }
    }

    __syncthreads();                        // all waves done with ks/vst
    if (it + 1 < kNT) lstore();             // waits on global loads here
    __syncthreads();
  }

  // normalize and write out [B, T, d]
  #pragma unroll
  for (int c = 0; c < 4; ++c) {
    const int hd = c * 16 + lm;
    #pragma unroll
    for (int i = 0; i < 8; ++i) {
      const int t = q0 + m0 + i + 8 * half;
      out[((size_t)b * kT + t) * kHD + hd] = oacc[c][i] / lrun[i];
    }
  }
}

// ---------------------------------------------------------------------------
extern "C" void kernel_launch(void* const* d_in, const int* in_sizes, int n_in,
                              void* d_out, int out_size, void* d_ws, size_t ws_size,
                              hipStream_t stream) {
  const float* x     = (const float*)d_in[0];
  const float* c_emb = (const float*)d_in[1];
  const float* Wq    = (const float*)d_in[2];
  const float* Wk    = (const float*)d_in[3];
  const float* Wv    = (const float*)d_in[4];
  const float* gamma = (const float*)d_in[5];
  const float* beta  = (const float*)d_in[6];
  float* out = (float*)d_out;

  float* q_buf = (float*)d_ws;                         // [B*T, 64]
  float* k_buf = q_buf + (size_t)kB * kT * kHD;        // [B, S, 64]
  float* v_buf = k_buf + (size_t)kB * kS * kHD;        // [B, S, 64]

  ln_mem_kernel<<<kB, 64, 0, stream>>>(c_emb, gamma, beta, k_buf, v_buf);
  qkv_kernel<<<(kB * kT) / 128, 256, 0, stream>>>(x, Wq, Wk, Wv,
                                                  q_buf, k_buf, v_buf);
  attn_kernel<<<dim3(kT / 128, kB), 256, 0, stream>>>(q_buf, k_buf, v_buf, out);
}